// KittenTTSTorch_2731599200577
// MI455X (gfx1250) — compile-verified
//
#include <hip/hip_runtime.h>
#include <cstdint>

// Problem constants from the reference
#define BB      8
#define TT      512
#define DD      512
#define MAXLEN  7680            // TT * MAX_DUR(15)
#define ROWV4   (DD / 4)        // 128 float4 per feature row (2 KB)

typedef float v4f __attribute__((ext_vector_type(4)));

// ---------------------------------------------------------------------------
// Kernel 1: per-batch inclusive scan of clamp(durations,1) -> cum[b,t]
// 8 blocks x 512 threads, Hillis-Steele scan in LDS. Tiny (4096 ints).
// ---------------------------------------------------------------------------
__global__ __launch_bounds__(TT) void lr_scan(const int* __restrict__ dur,
                                              int* __restrict__ cum) {
    __shared__ int s[TT];
    const int b = blockIdx.x;
    const int t = threadIdx.x;
    int d = dur[b * TT + t];
    d = (d < 1) ? 1 : d;                 // clamp(min=1) as in reference
    s[t] = d;
    __syncthreads();
    for (int off = 1; off < TT; off <<= 1) {
        int v = (t >= off) ? s[t - off] : 0;
        __syncthreads();
        s[t] += v;
        __syncthreads();
    }
    cum[b * TT + t] = s[t];
}

// ---------------------------------------------------------------------------
// Kernel 2: zero-fill ONLY the padded tail frames (n >= cum[b, T-1]).
// One float4 per thread across the whole output; valid frames exit early,
// so valid data is written exactly once (by lr_expand), pad exactly once here.
// ---------------------------------------------------------------------------
__global__ __launch_bounds__(256) void lr_zero_tail(const int* __restrict__ cum,
                                                    float* __restrict__ out) {
    const size_t i = (size_t)blockIdx.x * blockDim.x + threadIdx.x; // float4 index
    const size_t perB = (size_t)MAXLEN * ROWV4;
    const int b = (int)(i / perB);
    const int n = (int)((i - (size_t)b * perB) >> 7);               // frame index (/128)
    const int total = cum[b * TT + (TT - 1)];                       // valid length of batch b
    if (n >= total) {
        v4f z = {0.0f, 0.0f, 0.0f, 0.0f};
        __builtin_nontemporal_store(z, ((v4f*)out) + i);            // streaming NT b128 store
    }
}

// ---------------------------------------------------------------------------
// Kernel 3: expand. One 128-thread block per (b,t) phoneme.
// Stage the 2 KB source row in LDS via gfx1250 async load (read global ONCE),
// wait on ASYNCcnt, pull each lane's 16 B into registers (ds_load_b128),
// then stream it to dur[b,t] consecutive output frames with NT b128 stores.
// ---------------------------------------------------------------------------
__global__ __launch_bounds__(ROWV4) void lr_expand(const float* __restrict__ feat,
                                                   const int* __restrict__ cum,
                                                   float* __restrict__ out) {
    __shared__ v4f tile[ROWV4];                       // 2 KB staging buffer
    const int bt   = blockIdx.x;                      // b*TT + t
    const int b    = bt >> 9;                         // / 512
    const int t    = bt & (TT - 1);
    const int lane = threadIdx.x;                     // 0..127

    const int cend   = cum[bt];
    const int cstart = t ? cum[bt - 1] : 0;           // exclusive prefix within batch
    const int durv   = cend - cstart;                 // 1..15

    // Async copy: global feature row -> LDS (each lane moves its own 16 B).
    const v4f* gsrc = ((const v4f*)feat) + (size_t)bt * ROWV4 + lane;
    unsigned ldsoff = (unsigned)(uintptr_t)(&tile[lane]);   // low 32 bits = LDS offset
    asm volatile("global_load_async_to_lds_b128 %0, %1, off"
                 :: "v"(ldsoff), "v"(gsrc)
                 : "memory");
    // ASYNCcnt is per-wave; each lane only re-reads the bytes it loaded,
    // so a counter wait suffices (no workgroup barrier needed).
    asm volatile("s_wait_asynccnt 0" ::: "memory");

    const v4f val = tile[lane];                       // ds_load_b128, read once

    // Fan out: write the row to dur consecutive output frames.
    v4f* dst = ((v4f*)out) + ((size_t)b * MAXLEN + cstart) * ROWV4 + lane;
    for (int k = 0; k < durv; ++k) {
        __builtin_nontemporal_store(val, dst);        // streaming NT b128 store
        dst += ROWV4;
    }
}

// ---------------------------------------------------------------------------
// Launch wrapper
// ---------------------------------------------------------------------------
extern "C" void kernel_launch(void* const* d_in, const int* in_sizes, int n_in,
                              void* d_out, int out_size, void* d_ws, size_t ws_size,
                              hipStream_t stream) {
    const float* features  = (const float*)d_in[0];  // (B,T,D) f32
    const int*   durations = (const int*)d_in[1];    // (B,T)   i32
    float*       out       = (float*)d_out;          // (B,MAXLEN,D) f32
    int*         cum       = (int*)d_ws;             // 16 KB scratch

    // 1) per-batch inclusive scan of clamped durations
    lr_scan<<<BB, TT, 0, stream>>>(durations, cum);

    // 2) zero exactly the padded tail frames
    const size_t totalV4 = (size_t)BB * MAXLEN * ROWV4;       // 7,864,320
    const int    zblocks = (int)((totalV4 + 255) / 256);      // 30,720
    lr_zero_tail<<<zblocks, 256, 0, stream>>>(cum, out);

    // 3) expand valid frames (one block per phoneme)
    lr_expand<<<BB * TT, ROWV4, 0, stream>>>(features, cum, out);
}